// Linear_66142496358773
// MI455X (gfx1250) — compile-verified
//
#include <hip/hip_runtime.h>
#include <hip/hip_bf16.h>

// Problem constants (from reference)
#define BATCH     32
#define SEQ       128
#define IN_F      256
#define OUT_F     256
#define NUM_BASIS 1024
#define M_TOTAL   (BATCH * SEQ)        // 4096
#define WMIX_ELEMS (OUT_F * IN_F)      // 65536
#define SPLIT_K   16                   // basis-dim split for kernel 1
#define BASIS_PER_SLICE (NUM_BASIS / SPLIT_K)  // 64

typedef float v2f __attribute__((ext_vector_type(2)));
typedef float v8f __attribute__((ext_vector_type(8)));

// ---------------------------------------------------------------------------
// Kernel 1: partial[slice][j] = sum_{b in slice} alpha[b] * W[b][j]
// Pure HBM stream of 268 MB; float4 per thread -> 512B contiguous per wave
// per step. alpha[b] is wave-uniform -> scalar load path.
// grid = (64, SPLIT_K), block = 256
// ---------------------------------------------------------------------------
__global__ void __launch_bounds__(256)
basis_mix_partial(const float* __restrict__ W,
                  const float* __restrict__ alpha,
                  float* __restrict__ partial)
{
    const int j4    = blockIdx.x * 256 + threadIdx.x;   // float4 index into 65536 floats
    const int slice = blockIdx.y;
    const float4* __restrict__ W4 = (const float4*)W;

    float ax = 0.f, ay = 0.f, az = 0.f, aw = 0.f;
    const int b0 = slice * BASIS_PER_SLICE;

#pragma unroll 4
    for (int bb = 0; bb < BASIS_PER_SLICE; ++bb) {
        const int b = b0 + bb;
        const float a = alpha[b];                        // uniform -> SGPR
        const float4 w = W4[(size_t)b * (WMIX_ELEMS / 4) + j4];
        ax = fmaf(a, w.x, ax);
        ay = fmaf(a, w.y, ay);
        az = fmaf(a, w.z, az);
        aw = fmaf(a, w.w, aw);
    }
    float4 r; r.x = ax; r.y = ay; r.z = az; r.w = aw;
    ((float4*)partial)[(size_t)slice * (WMIX_ELEMS / 4) + j4] = r;
}

// ---------------------------------------------------------------------------
// Kernel 2: Wmix[j] = sum over SPLIT_K partials (fixed order -> deterministic)
// grid = 64, block = 256
// ---------------------------------------------------------------------------
__global__ void __launch_bounds__(256)
reduce_mix(const float* __restrict__ partial, float* __restrict__ Wmix)
{
    const int j4 = blockIdx.x * 256 + threadIdx.x;
    const float4* __restrict__ P4 = (const float4*)partial;
    float sx = 0.f, sy = 0.f, sz = 0.f, sw = 0.f;
#pragma unroll
    for (int sl = 0; sl < SPLIT_K; ++sl) {
        const float4 v = P4[(size_t)sl * (WMIX_ELEMS / 4) + j4];
        sx += v.x; sy += v.y; sz += v.z; sw += v.w;
    }
    float4 r; r.x = sx; r.y = sy; r.z = sz; r.w = sw;
    ((float4*)Wmix)[j4] = r;
}

// ---------------------------------------------------------------------------
// Kernel 3: out[m, o] = x[m, :] . Wmix[o, :] + bias[o]
// M=4096, N=256, K=256, all f32 -> V_WMMA_F32_16X16X4_F32.
// Block = 256 threads (8 waves). Block tile: 128 (M) x 16 (N).
// Wave w computes the 16x16 tile at rows [mBase + 16w, +16).
// K is chunked by 64 through LDS (padded pitch 68 floats: stride mod 64 = 4,
// so the 16-lane row fan-out hits distinct banks).
//
// ISA layouts used (CDNA5 ISA):
//   A 16x4 f32 (2 VGPRs): lanes 0-15 -> M=lane, K = {0,1}; lanes 16-31 -> K = {2,3}
//   B 4x16  (2 VGPRs): lane n = lane&15, K = {0,1} (+2 for lanes 16-31)
//   C/D 16x16 f32 (8 VGPRs): VGPR r -> M=r (lanes 0-15), M=r+8 (lanes 16-31), N=lane&15
// ---------------------------------------------------------------------------
#define KCHUNK 64
#define LDS_PITCH 68   // 64 + 4 pad floats

__global__ void __launch_bounds__(256)
mix_gemm_wmma(const float* __restrict__ x,
              const float* __restrict__ Wmix,
              const float* __restrict__ bias,
              float* __restrict__ out)
{
    __shared__ float As[128 * LDS_PITCH];  // 128 M-rows x KCHUNK
    __shared__ float Bs[16 * LDS_PITCH];   // 16 N-rows  x KCHUNK

    const int tid   = threadIdx.x;
    const int wave  = tid >> 5;
    const int lane  = tid & 31;
    const int mBase = blockIdx.y * 128;
    const int nBase = blockIdx.x * 16;

    v8f acc = {};  // 16x16 f32 accumulator (8 VGPRs)

    const int lrow16 = lane & 15;
    const int koff   = (lane >> 4) << 1;   // 0 for lanes 0-15, 2 for lanes 16-31

    for (int kc = 0; kc < IN_F; kc += KCHUNK) {
        // ---- stage x tile: 128 rows x 64 floats, float4-coalesced ----
#pragma unroll
        for (int it = 0; it < 8; ++it) {
            const int idx = tid + it * 256;      // 2048 float4 slots
            const int row = idx >> 4;            // /16 float4 per row
            const int kq  = (idx & 15) << 2;     // float offset in chunk
            const float4 v = *(const float4*)(x + (size_t)(mBase + row) * IN_F + kc + kq);
            *(float4*)(&As[row * LDS_PITCH + kq]) = v;
        }
        // ---- stage Wmix tile: 16 rows x 64 floats ----
        {
            const int row = tid >> 4;
            const int kq  = (tid & 15) << 2;
            const float4 v = *(const float4*)(Wmix + (size_t)(nBase + row) * IN_F + kc + kq);
            *(float4*)(&Bs[row * LDS_PITCH + kq]) = v;
        }
        __syncthreads();

        const float* aRow = &As[(wave * 16 + lrow16) * LDS_PITCH];
        const float* bRow = &Bs[lrow16 * LDS_PITCH];

#pragma unroll
        for (int k0 = 0; k0 < KCHUNK; k0 += 4) {
            const v2f a = *(const v2f*)(aRow + k0 + koff);
            const v2f b = *(const v2f*)(bRow + k0 + koff);
            acc = __builtin_amdgcn_wmma_f32_16x16x4_f32(
                /*neg_a=*/false, a, /*neg_b=*/false, b,
                /*c_mod=*/(short)0, acc, /*reuse_a=*/false, /*reuse_b=*/false);
        }
        __syncthreads();
    }

    // ---- epilogue: fused bias add + store ----
    const int col  = nBase + lrow16;
    const float bv = bias[col];
    const int mTop = mBase + (wave << 4) + ((lane >> 4) << 3);  // +8 rows for hi lanes
#pragma unroll
    for (int r = 0; r < 8; ++r) {
        out[(size_t)(mTop + r) * OUT_F + col] = acc[r] + bv;
    }
}

// ---------------------------------------------------------------------------
// Launch. d_in order: x, alpha, bias, W, seed.
// Workspace layout: [ partial: SPLIT_K * 65536 f32 (4 MB) | Wmix: 65536 f32 ]
// ---------------------------------------------------------------------------
extern "C" void kernel_launch(void* const* d_in, const int* in_sizes, int n_in,
                              void* d_out, int out_size, void* d_ws, size_t ws_size,
                              hipStream_t stream) {
    const float* x     = (const float*)d_in[0];
    const float* alpha = (const float*)d_in[1];
    const float* bias  = (const float*)d_in[2];
    const float* W     = (const float*)d_in[3];
    float* out = (float*)d_out;

    float* partial = (float*)d_ws;
    float* Wmix    = partial + (size_t)SPLIT_K * WMIX_ELEMS;

    // Kernel 1: 268 MB stream, 1024 blocks for bandwidth saturation
    dim3 g1(WMIX_ELEMS / (256 * 4), SPLIT_K);
    basis_mix_partial<<<g1, 256, 0, stream>>>(W, alpha, partial);

    // Kernel 2: fixed-order reduction
    reduce_mix<<<WMIX_ELEMS / (256 * 4), 256, 0, stream>>>(partial, Wmix);

    // Kernel 3: WMMA GEMM, grid = (N tiles, M super-tiles) = (16, 32)
    dim3 g3(OUT_F / 16, M_TOTAL / 128);
    mix_gemm_wmma<<<g3, 256, 0, stream>>>(x, Wmix, bias, out);
}